// FeedForwardNet_Pool_83648783057346
// MI455X (gfx1250) — compile-verified
//
#include <hip/hip_runtime.h>

// Depthwise 1-D conv, 13 asymmetric taps shared across channels.
// x: (32,512,4096) f32  -> rows = 16384, L = 4096
// out[q] = sum_{j=0..12} kern[j] * x[q - 6 + j], kern = [w6..w0, w7..w12]
//
// Design (MI455X, HBM-bound ~22us floor):
//  - each 256-thread block owns half a row (2048 outputs)
//  - async global->LDS b128 copies fill a 2064-float tile (8-elem halo each side)
//  - each wave computes 256 outputs as Out(16x16) = Xext(16x28) x Kband(28x16)
//    via 7 chained V_WMMA_F32_16X16X4_F32 (K = 28 = 7*4)

#define KHALF   7
#define TAPS    13
#define ROWLEN  4096
#define NROWS   16384
#define TILE    2048
#define HALO    8
#define LDSN    (TILE + 2 * HALO)   // 2064 floats
#define NCHUNK  (LDSN / 4)          // 516 float4 chunks

typedef __attribute__((ext_vector_type(2))) float v2f;
typedef __attribute__((ext_vector_type(8))) float v8f;
typedef __attribute__((ext_vector_type(4))) int   v4i;

typedef __attribute__((address_space(1))) v4i g_v4i;
typedef __attribute__((address_space(3))) v4i l_v4i;

#if __has_builtin(__builtin_amdgcn_global_load_async_to_lds_b128)
#define HAVE_ASYNC_LDS 1
#endif

__global__ __launch_bounds__(256) void conv13_wmma(const float* __restrict__ x,
                                                   const float* __restrict__ w,
                                                   float* __restrict__ out) {
  __shared__ __align__(16) float tile[LDSN];
  __shared__ float kw[TAPS + 3];

  const int tid  = threadIdx.x;
  const int row  = (int)blockIdx.x >> 1;
  const int h    = (int)blockIdx.x & 1;           // which half of the row
  const int base = row * ROWLEN + h * TILE;       // first output element index
  const int g0   = base - HALO;                   // global element index of tile[0]

  // Stage cross-correlation taps in LDS: kern[j] = j<7 ? w[6-j] : w[j]
  if (tid < TAPS) kw[tid] = (tid < KHALF) ? w[KHALF - 1 - tid] : w[tid];

  // Fill the LDS tile. Out-of-row regions are exactly the first two /
  // last two 16B chunks of the tile (halo==8 aligns with float4).
  for (int i = tid; i < NCHUNK; i += 256) {
    const bool zc = (h == 0) ? (i < 2) : (i >= NCHUNK - 2);
    if (zc) {
      float4 z = {0.f, 0.f, 0.f, 0.f};
      *(float4*)(&tile[4 * i]) = z;
    } else {
#ifdef HAVE_ASYNC_LDS
      // ROCm convention (cf. __builtin_amdgcn_global_load_lds):
      //   (global src, LDS dst, imm offset, imm cpol)
      __builtin_amdgcn_global_load_async_to_lds_b128(
          (g_v4i*)const_cast<float*>(x + (g0 + 4 * i)),
          (l_v4i*)(&tile[4 * i]),
          0, 0);
#else
      *(float4*)(&tile[4 * i]) = *(const float4*)(x + (g0 + 4 * i));
#endif
    }
  }

#ifdef HAVE_ASYNC_LDS
#if __has_builtin(__builtin_amdgcn_s_wait_asynccnt)
  __builtin_amdgcn_s_wait_asynccnt(0);
#else
  asm volatile("s_wait_asynccnt 0" ::: "memory");
#endif
#endif
  __syncthreads();

  const int lane = tid & 31;     // wave32
  const int wv   = tid >> 5;     // 8 waves, 256 outputs each
  const int n    = lane & 15;    // N column (and M row for the A operand)
  const int hf   = lane >> 4;    // lane half selects K pair

#if __has_builtin(__builtin_amdgcn_wmma_f32_16x16x4_f32)
  // Banded kernel matrix operand: B_t[k][n] with k = 2*hf + {0,1}, K-slice 4t.
  v2f bv[7];
#pragma unroll
  for (int t = 0; t < 7; ++t) {
    const int p0 = 4 * t + 2 * hf;
    const int d0 = p0 - n;
    const int d1 = d0 + 1;
    bv[t].x = (d0 >= 0 && d0 < TAPS) ? kw[d0] : 0.f;
    bv[t].y = (d1 >= 0 && d1 < TAPS) ? kw[d1] : 0.f;
  }

  // Xext[M][p] = tile[256*wv + 16*M + p + 2]; A lane M = n, K = 2*hf + {0,1}.
  // Consecutive K -> one aligned 8-byte LDS load per WMMA step.
  v8f acc = {0.f, 0.f, 0.f, 0.f, 0.f, 0.f, 0.f, 0.f};
  const v2f* t2 = (const v2f*)tile;
#pragma unroll
  for (int t = 0; t < 7; ++t) {
    v2f a = t2[128 * wv + 8 * n + 2 * t + hf + 1];
    acc = __builtin_amdgcn_wmma_f32_16x16x4_f32(false, a, false, bv[t],
                                                (short)0, acc, false, false);
  }

  // D layout: VGPR r -> M = r + 8*hf, N = n.  Output pos = 256*wv + 16*M + N.
#pragma unroll
  for (int r = 0; r < 8; ++r) {
    out[base + 256 * wv + 16 * r + 128 * hf + n] = acc[r];
  }
#else
  // VALU fallback (same output mapping)
#pragma unroll
  for (int r = 0; r < 8; ++r) {
    const int q = 256 * wv + 16 * r + 128 * hf + n;
    float s = 0.f;
#pragma unroll
    for (int j = 0; j < TAPS; ++j) s += kw[j] * tile[q + 2 + j];
    out[base + q] = s;
  }
#endif
}

extern "C" void kernel_launch(void* const* d_in, const int* in_sizes, int n_in,
                              void* d_out, int out_size, void* d_ws, size_t ws_size,
                              hipStream_t stream) {
  const float* x = (const float*)d_in[0];        // (32,512,4096) f32
  const float* w = (const float*)d_in[1];        // (1,1,13) f32
  float* out = (float*)d_out;                    // (32,512,4096) f32
  conv13_wmma<<<dim3(NROWS * 2), dim3(256), 0, stream>>>(x, w, out);
}